// GaussianSQuantizer_33191507264266
// MI455X (gfx1250) — compile-verified
//
#include <hip/hip_runtime.h>
#include <math.h>

typedef __attribute__((ext_vector_type(2))) float v2f;
typedef __attribute__((ext_vector_type(8))) float v8f;
typedef int v4i_vs __attribute__((vector_size(16)));   // matches builtin's pointee type

#define N_Z    65536
#define NE     4096
#define E_DIM  128
#define MTILE  128
#define NTILE  64
#define NTILES (NE / NTILE)
#define ZSTR   132   // 128 + 4 pad: 132 % 64 == 4 -> conflict-free WMMA operand loads
#define ESTR   132
#define LSTR   66    // 64 + 2 pad: conflict-free logit writes/reads
#define LOG2E_F   1.4426950408889634f
#define LOG2PI_F  1.8378770664093453f

#define AS1 __attribute__((address_space(1)))
#define AS3 __attribute__((address_space(3)))

#if __has_builtin(__builtin_amdgcn_global_load_async_to_lds_b128) && \
    __has_builtin(__builtin_amdgcn_s_wait_asynccnt)
#define HAS_ASYNC_LDS 1
#else
#define HAS_ASYNC_LDS 0
#endif

__device__ __forceinline__ float fast_exp(float x) {
    return __builtin_amdgcn_exp2f(x * LOG2E_F);
}

// Stage one 64x128 f32 embedding tile (2048 float4) into LDS.
__device__ __forceinline__ void etile_load_async(const float4* gsrc, float* ldst, int t) {
#if HAS_ASYNC_LDS
#pragma unroll
    for (int j = 0; j < 8; ++j) {
        int flat = t + 256 * j;
        int r  = flat >> 5;
        int c4 = flat & 31;
        __builtin_amdgcn_global_load_async_to_lds_b128(
            (AS1 v4i_vs*)(gsrc + (size_t)r * 32 + c4),
            (AS3 v4i_vs*)&ldst[r * ESTR + c4 * 4],
            0, 0);
    }
#endif
}

__device__ __forceinline__ void etile_load_sync(const float4* gsrc, float* ldst, int t) {
#pragma unroll
    for (int j = 0; j < 8; ++j) {
        int flat = t + 256 * j;
        int r  = flat >> 5;
        int c4 = flat & 31;
        float4 q = gsrc[(size_t)r * 32 + c4];
        *(float4*)&ldst[r * ESTR + c4 * 4] = q;
    }
}

// ---- Kernel 1: halfe2[k] = 0.5 * ||embedding[k]||^2 ; zero loss accumulators ----
__global__ void gsq_prep(const float* __restrict__ emb,
                         float* __restrict__ halfe2,
                         float* __restrict__ loss_out) {
    int k = blockIdx.x * blockDim.x + threadIdx.x;
    if (k < NE) {
        const float4* row = (const float4*)(emb + (size_t)k * E_DIM);
        float s = 0.f;
#pragma unroll
        for (int j = 0; j < E_DIM / 4; ++j) {
            float4 q = row[j];
            s += q.x * q.x + q.y * q.y + q.z * q.z + q.w * q.w;
        }
        halfe2[k] = 0.5f * s;
    }
    if (k == 0) { loss_out[0] = 0.f; loss_out[1] = 0.f; }
}

// ---- Kernel 2: fused GEMM(WMMA f32) + online softmax + argmax + gather ----
__global__ __launch_bounds__(256) void gsq_main(
    const float* __restrict__ z, const float* __restrict__ emb,
    const float* __restrict__ var_q_logit, const float* __restrict__ var_init,
    const float* __restrict__ halfe2,
    float* __restrict__ zq_out, float* __restrict__ loss_out)
{
    __shared__ float z_lds[MTILE * ZSTR];
    __shared__ float e_lds[2][NTILE * ESTR];   // double-buffered async staging
    __shared__ float l_lds[MTILE * LSTR];
    __shared__ float sz_lds[MTILE];
    __shared__ float se_lds[NTILE];
    __shared__ int   bidx_lds[MTILE];
    __shared__ float redA[MTILE];
    __shared__ float redB[MTILE];

    const int t    = threadIdx.x;
    const int lane = t & 31;          // wave32
    const int wave = t >> 5;          // 8 waves
    const int lm   = lane & 15;
    const int hh   = lane >> 4;       // half-wave selector
    const int row_base = blockIdx.x * MTILE;

    // scalar parameters (global var mode)
    const float lgt   = var_q_logit[0];
    const float v     = (1.f / (1.f + fast_exp(-lgt))) * 2.f * var_init[0];
    const float inv_v = 1.f / v;
    const float Kc    = 0.5f * (float)E_DIM * (__logf(v) + LOG2PI_F);

#if HAS_ASYNC_LDS
    // prologue: kick off tile 0 while we stage z
    etile_load_async((const float4*)emb, e_lds[0], t);
#endif

    // ---- stage z tile: 128 rows x 128 f32, coalesced float4 ----
    {
        const float4* zg = (const float4*)(z + (size_t)row_base * E_DIM);
#pragma unroll
        for (int j = 0; j < 16; ++j) {
            int flat = t + 256 * j;        // 4096 float4
            int r  = flat >> 5;
            int c4 = flat & 31;
            float4 q = zg[r * 32 + c4];
            *(float4*)&z_lds[r * ZSTR + c4 * 4] = q;
        }
    }
    __syncthreads();

    // ---- 0.5*||z_row||^2 (2 threads per row, shuffle-merge) ----
    {
        int r = t >> 1, half = t & 1;
        const float* p = &z_lds[r * ZSTR + half * 64];
        float s = 0.f;
#pragma unroll
        for (int j = 0; j < 64; ++j) s += p[j] * p[j];
        s += __shfl_xor(s, 1, 32);
        if (half == 0) sz_lds[r] = 0.5f * s;
    }

    // online softmax + argmax state; thread t owns row t/2, cols (t&1)*32..+32
    const int r_red = t >> 1, half_red = t & 1;
    float run_m = -__builtin_inff(), run_s = 0.f, run_t = 0.f;
    float best_v = -__builtin_inff(); int best_i = 0;

    for (int nt = 0; nt < NTILES; ++nt) {
        const int cur = nt & 1;
        const float* ecur;
#if HAS_ASYNC_LDS
        ecur = e_lds[cur];
        if (nt + 1 < NTILES) {
            // prefetch next tile into the other buffer, then drain all but it
            etile_load_async((const float4*)(emb + (size_t)(nt + 1) * NTILE * E_DIM),
                             e_lds[cur ^ 1], t);
            __builtin_amdgcn_s_wait_asynccnt(1);
        } else {
            __builtin_amdgcn_s_wait_asynccnt(0);
        }
#else
        ecur = e_lds[0];
        if (nt + 1 < NTILES)   // hint next tile toward the caches
            __builtin_prefetch(emb + (size_t)(nt + 1) * NTILE * E_DIM + t * 32, 0, 1);
        etile_load_sync((const float4*)(emb + (size_t)nt * NTILE * E_DIM), e_lds[0], t);
#endif
        if (t < NTILE) se_lds[t] = halfe2[nt * NTILE + t];
        __syncthreads();

        // ---- WMMA GEMM: wave computes rows [wave*16, wave*16+16) x 64 codes ----
        {
            v8f acc0 = {}, acc1 = {}, acc2 = {}, acc3 = {};
            const int m0 = wave * 16;
            // A layout: lanes 0-15 -> M=lane, VGPR{0,1}=K{0,1}; lanes 16-31 -> K{2,3}
            const float* arow = &z_lds[(m0 + lm) * ZSTR + 2 * hh];
            const float* b0 = &ecur[( 0 + lm) * ESTR + 2 * hh];
            const float* b1 = &ecur[(16 + lm) * ESTR + 2 * hh];
            const float* b2 = &ecur[(32 + lm) * ESTR + 2 * hh];
            const float* b3 = &ecur[(48 + lm) * ESTR + 2 * hh];
#pragma unroll
            for (int kk = 0; kk < E_DIM; kk += 4) {
                v2f a  = *(const v2f*)(arow + kk);
                v2f q0 = *(const v2f*)(b0 + kk);
                v2f q1 = *(const v2f*)(b1 + kk);
                v2f q2 = *(const v2f*)(b2 + kk);
                v2f q3 = *(const v2f*)(b3 + kk);
                acc0 = __builtin_amdgcn_wmma_f32_16x16x4_f32(false, a, false, q0, (short)0, acc0, false, false);
                acc1 = __builtin_amdgcn_wmma_f32_16x16x4_f32(false, a, false, q1, (short)0, acc1, false, false);
                acc2 = __builtin_amdgcn_wmma_f32_16x16x4_f32(false, a, false, q2, (short)0, acc2, false, false);
                acc3 = __builtin_amdgcn_wmma_f32_16x16x4_f32(false, a, false, q3, (short)0, acc3, false, false);
            }
            // C layout: VGPR i -> row m0 + i + 8*hh, col lm (per 16-col subtile)
#pragma unroll
            for (int i = 0; i < 8; ++i) {
                int   rr   = m0 + i + 8 * hh;
                float sz_v = sz_lds[rr];
                float* lrow = &l_lds[rr * LSTR];
                lrow[ 0 + lm] = (acc0[i] - sz_v - se_lds[ 0 + lm]) * inv_v - Kc;
                lrow[16 + lm] = (acc1[i] - sz_v - se_lds[16 + lm]) * inv_v - Kc;
                lrow[32 + lm] = (acc2[i] - sz_v - se_lds[32 + lm]) * inv_v - Kc;
                lrow[48 + lm] = (acc3[i] - sz_v - se_lds[48 + lm]) * inv_v - Kc;
            }
        }
        __syncthreads();

        // ---- online softmax / argmax update over this tile ----
        {
            const float* lp = &l_lds[r_red * LSTR + half_red * 32];
            const int code0 = nt * NTILE + half_red * 32;
#pragma unroll
            for (int c = 0; c < 32; ++c) {
                float L = lp[c];
                if (L > best_v) { best_v = L; best_i = code0 + c; }
                if (L > run_m) {
                    float sc = fast_exp(run_m - L);   // exp2(-inf)=0 handles first hit
                    run_s = run_s * sc + 1.f;
                    run_t = run_t * sc + L;
                    run_m = L;
                } else {
                    float e = fast_exp(L - run_m);
                    run_s += e;
                    run_t += e * L;
                }
            }
        }
        __syncthreads();   // also fences e_lds[cur] before it is refilled at nt+2
    }

    // ---- merge the two half-row states (adjacent lanes, same wave) ----
    {
        float om  = __shfl_xor(run_m, 1, 32);
        float os  = __shfl_xor(run_s, 1, 32);
        float ot  = __shfl_xor(run_t, 1, 32);
        float obv = __shfl_xor(best_v, 1, 32);
        int   obi = __shfl_xor(best_i, 1, 32);
        float M  = fmaxf(run_m, om);
        float c1 = fast_exp(run_m - M), c2 = fast_exp(om - M);
        float S  = run_s * c1 + os * c2;
        float T  = run_t * c1 + ot * c2;
        if (obv > best_v) { best_v = obv; best_i = obi; }
        if (half_red == 0) {
            float pl = T / S;                       // sum p * logit
            redA[r_red] = pl - (M + __logf(S));     // sum p * log_p
            redB[r_red] = -pl;                      // commit
            bidx_lds[r_red] = best_i;
        }
    }
    __syncthreads();

    // ---- block reduction of per-row loss contributions ----
    for (int off = 64; off > 0; off >>= 1) {
        if (t < off) { redA[t] += redA[t + off]; redB[t] += redB[t + off]; }
        __syncthreads();
    }
    if (t == 0) {
        atomicAdd(&loss_out[0], redA[0] * (1.f / (float)N_Z));
        atomicAdd(&loss_out[1], redB[0] * (1.f / (float)N_Z));
    }

    // ---- coalesced gather: z_quantized[row] = embedding[argmax] ----
#pragma unroll
    for (int j = 0; j < 16; ++j) {
        int flat = t + 256 * j;       // 4096 float4; one wave streams one row
        int r  = flat >> 5;
        int c4 = flat & 31;
        int code = bidx_lds[r];
        float4 q = ((const float4*)(emb + (size_t)code * E_DIM))[c4];
        ((float4*)(zq_out + (size_t)(row_base + r) * E_DIM))[c4] = q;
    }
}

extern "C" void kernel_launch(void* const* d_in, const int* in_sizes, int n_in,
                              void* d_out, int out_size, void* d_ws, size_t ws_size,
                              hipStream_t stream) {
    const float* z   = (const float*)d_in[0];
    const float* emb = (const float*)d_in[1];
    const float* vql = (const float*)d_in[2];
    const float* vin = (const float*)d_in[3];
    float* out    = (float*)d_out;
    float* loss   = out + (size_t)N_Z * E_DIM;   // [kld, commit] after z_quantized
    float* halfe2 = (float*)d_ws;                // 4096 floats scratch

    gsq_prep<<<NE / 256, 256, 0, stream>>>(emb, halfe2, loss);
    gsq_main<<<N_Z / MTILE, 256, 0, stream>>>(z, emb, vql, vin, halfe2, out, loss);
}